// GraphSAGE_39874476376471
// MI455X (gfx1250) — compile-verified
//
#include <hip/hip_runtime.h>

typedef float v2f __attribute__((ext_vector_type(2)));
typedef float v8f __attribute__((ext_vector_type(8)));

#define DIM 64
#define BSTRIDE 80   // float2 row stride for LDS W tiles (bank-conflict padding)

// ---------------------------------------------------------------------------
// Zero-fill kernel (workspace accumulators must be zeroed every call).
// ---------------------------------------------------------------------------
__global__ __launch_bounds__(256) void sage_zero(float* __restrict__ p, int n) {
    int i = blockIdx.x * 256 + threadIdx.x;
    if (i < n) p[i] = 0.0f;
}

// ---------------------------------------------------------------------------
// Degree count: cnt[dst[e]] += 1  (same for both layers, computed once).
// ---------------------------------------------------------------------------
__global__ __launch_bounds__(256) void sage_count(const int* __restrict__ dst,
                                                  float* __restrict__ cnt, int n_edges) {
    int e = blockIdx.x * 256 + threadIdx.x;
    if (e >= n_edges) return;
    __hip_atomic_fetch_add(&cnt[dst[e]], 1.0f, __ATOMIC_RELAXED, __HIP_MEMORY_SCOPE_AGENT);
}

// ---------------------------------------------------------------------------
// Edge aggregation: agg[dst] += x[src].  16 threads per edge, float4 chunks.
// x rows (25.6MB) and agg (25.6MB) both live in the 192MB L2, so the gather
// and the f32 atomics are L2-resident; HBM only sees the edge list.
// ---------------------------------------------------------------------------
__global__ __launch_bounds__(256) void sage_aggregate(const float* __restrict__ x,
                                                      const int* __restrict__ src,
                                                      const int* __restrict__ dst,
                                                      float* __restrict__ agg, int n_edges) {
    int t = blockIdx.x * 256 + threadIdx.x;
    int e = t >> 4;
    if (e >= n_edges) return;
    int c = (t & 15) * 4;
    int s = src[e];
    int d = dst[e];
    const float4 v = *reinterpret_cast<const float4*>(x + (size_t)s * DIM + c);
    float* a = agg + (size_t)d * DIM + c;
    __hip_atomic_fetch_add(a + 0, v.x, __ATOMIC_RELAXED, __HIP_MEMORY_SCOPE_AGENT);
    __hip_atomic_fetch_add(a + 1, v.y, __ATOMIC_RELAXED, __HIP_MEMORY_SCOPE_AGENT);
    __hip_atomic_fetch_add(a + 2, v.z, __ATOMIC_RELAXED, __HIP_MEMORY_SCOPE_AGENT);
    __hip_atomic_fetch_add(a + 3, v.w, __ATOMIC_RELAXED, __HIP_MEMORY_SCOPE_AGENT);
}

// ---------------------------------------------------------------------------
// Fused mean + dual-GEMM + bias (+ReLU):
//   out = (agg * 1/max(cnt,1)) @ W_l + x @ W_r + b
// One wave per 16-node tile; 128 x v_wmma_f32_16x16x4_f32 per tile (exact f32).
//
// W is staged in LDS as row-pair-interleaved float2 with padded stride:
//   sW[p*BSTRIDE + n] = { W[2p][n], W[2p+1][n] }
// so each B operand is one aligned ds_load_b64 (no repack movs), and the
// BSTRIDE=80 padding puts the two half-waves (pair-rows 2k vs 2k+1) on
// disjoint 32-bank sets -> conflict-free 64-bank b64 loads.
//
// A layout : lanes 0-15 -> K = {4k,4k+1}; lanes 16-31 -> K = {4k+2,4k+3}
// B layout : vgpr j, lane L -> W[4k + 2*(L/16) + j][16*nt + L%16]
// C layout : vgpr v, lane L -> out[M = v + 8*(L/16)][16*nt + L%16]
// ---------------------------------------------------------------------------
__global__ __launch_bounds__(256) void sage_gemm(const float* __restrict__ agg,
                                                 const float* __restrict__ cnt,
                                                 const float* __restrict__ xin,
                                                 const float* __restrict__ Wl,
                                                 const float* __restrict__ Wr,
                                                 const float* __restrict__ bias,
                                                 float* __restrict__ out,
                                                 int n_nodes, int do_relu) {
    __shared__ v2f sWl[32 * BSTRIDE];
    __shared__ v2f sWr[32 * BSTRIDE];
    int tid = threadIdx.x;
    {
        float* fl = (float*)sWl;
        float* fr = (float*)sWr;
        for (int i = tid; i < DIM * DIM; i += 256) {
            int k = i >> 6;          // source row
            int n = i & 63;          // source col
            int idx = ((k >> 1) * BSTRIDE + n) * 2 + (k & 1);
            fl[idx] = Wl[i];
            fr[idx] = Wr[i];
        }
    }
    __syncthreads();

    int wave = tid >> 5;   // 8 waves per block, one 16-row tile per wave
    int lane = tid & 31;
    int tile = blockIdx.x * 8 + wave;
    if (tile * 16 >= n_nodes) return;   // wave-uniform exit (EXEC stays all-ones for WMMA)

    int row = lane & 15;   // M row within tile (A) / N column within tile (B,C)
    int hi  = lane >> 4;   // selects K sub-pair for A/B, M+8 for C
    int node = tile * 16 + row;

    float inv = 1.0f / fmaxf(cnt[node], 1.0f);
    const float* aggRow = agg + (size_t)node * DIM + 2 * hi;
    const float* xRow   = xin + (size_t)node * DIM + 2 * hi;

    // Per-lane base into the pair-interleaved W tiles: pair-row hi, column `row`
    const v2f* blBase = sWl + (size_t)hi * BSTRIDE + row;
    const v2f* brBase = sWr + (size_t)hi * BSTRIDE + row;

    v8f acc[4];
    for (int nt = 0; nt < 4; ++nt) {
        float bv = bias[nt * 16 + row];
        for (int v = 0; v < 8; ++v) acc[nt][v] = bv;
    }

    for (int kk = 0; kk < 16; ++kk) {
        int kbase = kk * 4;
        v2f am = *reinterpret_cast<const v2f*>(aggRow + kbase);
        v2f ax = *reinterpret_cast<const v2f*>(xRow + kbase);
        am = am * inv;
        int poff = 2 * kk * BSTRIDE;     // uniform pair-row offset for this K-step
        for (int nt = 0; nt < 4; ++nt) {
            v2f bl = blBase[poff + nt * 16];
            v2f br = brBase[poff + nt * 16];
            acc[nt] = __builtin_amdgcn_wmma_f32_16x16x4_f32(
                false, am, false, bl, (short)0, acc[nt], false, false);
            acc[nt] = __builtin_amdgcn_wmma_f32_16x16x4_f32(
                false, ax, false, br, (short)0, acc[nt], false, false);
        }
    }

    for (int nt = 0; nt < 4; ++nt) {
        int ncol = nt * 16 + row;
        for (int v = 0; v < 8; ++v) {
            int m = v + 8 * hi;
            float val = acc[nt][v];
            if (do_relu) val = fmaxf(val, 0.0f);
            out[(size_t)(tile * 16 + m) * DIM + ncol] = val;
        }
    }
}

// ---------------------------------------------------------------------------
// Launch: zero -> count -> aggregate(x) -> gemm(relu) -> zero -> aggregate(h)
//         -> gemm(out).  All async on `stream`; workspace re-initialized each
//         call for determinism.
// ---------------------------------------------------------------------------
extern "C" void kernel_launch(void* const* d_in, const int* in_sizes, int n_in,
                              void* d_out, int out_size, void* d_ws, size_t ws_size,
                              hipStream_t stream) {
    const float* x   = (const float*)d_in[0];
    const int*   ei  = (const int*)d_in[1];   // [2, E] flattened
    const float* Wl1 = (const float*)d_in[2];
    const float* Wr1 = (const float*)d_in[3];
    const float* b1  = (const float*)d_in[4];
    const float* Wl2 = (const float*)d_in[5];
    const float* Wr2 = (const float*)d_in[6];
    const float* b2  = (const float*)d_in[7];
    float* out = (float*)d_out;

    const int N = in_sizes[0] / DIM;     // 100000
    const int E = in_sizes[1] / 2;       // 1600000
    const int* src = ei;
    const int* dst = ei + E;

    // Workspace carve: agg[N*64] | cnt[N] (contiguous with agg for one zero pass) | h[N*64]
    float* agg = (float*)d_ws;
    size_t aggElems = (size_t)N * DIM;
    float* cnt = agg + aggElems;
    size_t cntPad = ((size_t)N + 63) & ~(size_t)63;
    float* h = cnt + cntPad;

    const int zeroAll = (int)(aggElems + N);          // agg + cnt in one pass
    const int zeroAgg = (int)aggElems;
    dim3 blk(256);
    dim3 gZeroAll((zeroAll + 255) / 256);
    dim3 gZeroAgg((zeroAgg + 255) / 256);
    dim3 gCount((E + 255) / 256);
    dim3 gAggr(((size_t)E * 16 + 255) / 256);
    int ntiles = (N + 15) / 16;
    dim3 gGemm((ntiles + 7) / 8);

    // ---- Layer 1 ----
    sage_zero<<<gZeroAll, blk, 0, stream>>>(agg, zeroAll);
    sage_count<<<gCount, blk, 0, stream>>>(dst, cnt, E);
    sage_aggregate<<<gAggr, blk, 0, stream>>>(x, src, dst, agg, E);
    sage_gemm<<<gGemm, blk, 0, stream>>>(agg, cnt, x, Wl1, Wr1, b1, h, N, 1);

    // ---- Layer 2 ----
    sage_zero<<<gZeroAgg, blk, 0, stream>>>(agg, zeroAgg);
    sage_aggregate<<<gAggr, blk, 0, stream>>>(h, src, dst, agg, E);
    sage_gemm<<<gGemm, blk, 0, stream>>>(agg, cnt, h, Wl2, Wr2, b2, out, N, 0);
}